// PortfolioConstraintLayer_86157043958058
// MI455X (gfx1250) — compile-verified
//
#include <hip/hip_runtime.h>
#include <hip/hip_bf16.h>
#include <stdint.h>

typedef __attribute__((ext_vector_type(4))) float v4f;

#define NCOLS 2048
#define TPB   256
#define EPT   8                                  // elements per thread
#define VERY_NEG (-8.5070591730234616e37f)       // finfo(float32).min / 4

// Block-wide sum of two lanes of data. All threads return the full sums.
// wave32 butterfly + 8-wave LDS combine.
__device__ __forceinline__ void block_reduce2(float& a, float& b, float* sred, int tid) {
#pragma unroll
    for (int off = 16; off > 0; off >>= 1) {
        a += __shfl_xor(a, off, 32);
        b += __shfl_xor(b, off, 32);
    }
    const int wave = tid >> 5;
    __syncthreads();                              // WAR guard on sred reuse
    if ((tid & 31) == 0) { sred[2 * wave] = a; sred[2 * wave + 1] = b; }
    __syncthreads();
    a = 0.0f; b = 0.0f;
#pragma unroll
    for (int w = 0; w < TPB / 32; ++w) { a += sred[2 * w]; b += sred[2 * w + 1]; }
}

__global__ __launch_bounds__(TPB) void sparsemax_rows(
    const float* __restrict__ logits,
    const uint8_t* __restrict__ mask,
    float* __restrict__ out)
{
    __shared__ float zrow[NCOLS];                 // 8 KB row staging
    __shared__ float sred[2 * (TPB / 32)];

    const int tid = threadIdx.x;
    const long long base = (long long)blockIdx.x * NCOLS;

    // ---- CDNA5 async DMA: row of logits -> LDS (2 x b128 per thread) ----
    {
        const float* g0 = logits + base + tid * 4;        // chunks [0,256)
        const float* g1 = g0 + NCOLS / 2;                 // chunks [256,512)
        uint32_t l0 = (uint32_t)(uintptr_t)(&zrow[tid * 4]);  // low32 of flat = LDS offset
        uint32_t l1 = l0 + (NCOLS / 2) * 4;
        asm volatile("global_load_async_to_lds_b128 %0, %1, off"
                     :: "v"(l0), "v"(g0) : "memory");
        asm volatile("global_load_async_to_lds_b128 %0, %1, off"
                     :: "v"(l1), "v"(g1) : "memory");
    }

    // mask bytes for this thread's 8 elements (coalesced 8B loads, overlaps DMA)
    const uint32_t* mp = (const uint32_t*)(mask + base) + tid * 2;
    const uint32_t mlo = mp[0];
    const uint32_t mhi = mp[1];

#if __has_builtin(__builtin_amdgcn_s_wait_asynccnt)
    __builtin_amdgcn_s_wait_asynccnt(0);
#else
    asm volatile("s_wait_asynccnt 0x0" ::: "memory");
#endif
    __syncthreads();   // each wave drained its ASYNCcnt; barrier publishes LDS to all waves

    // ---- build masked z in registers; seed support = masked-in set ----
    float z[EPT];
    float s = 0.0f, c = 0.0f;
#pragma unroll
    for (int j = 0; j < EPT; ++j) {
        const uint32_t mbyte = (j < 4) ? ((mlo >> (8 * j)) & 0xFFu)
                                       : ((mhi >> (8 * (j - 4))) & 0xFFu);
        const float zv = zrow[tid * EPT + j];
        const bool act = (mbyte != 0u);
        z[j] = act ? zv : VERY_NEG;
        if (act) { s += zv; c += 1.0f; }
    }
    block_reduce2(s, c, sred, tid);

    if (c < 0.5f) {                               // fully masked row -> zeros
        v4f zv4 = {0.0f, 0.0f, 0.0f, 0.0f};
        __builtin_nontemporal_store(zv4, (v4f*)(out + base + tid * EPT));
        __builtin_nontemporal_store(zv4, (v4f*)(out + base + tid * EPT + 4));
        return;
    }

    // ---- Michelot fixed point: tau = (sum_{z>tau} z - 1) / |{z>tau}| ----
    float tau   = (s - 1.0f) / c;
    float prevc = c;
    for (;;) {
        s = 0.0f; c = 0.0f;
#pragma unroll
        for (int j = 0; j < EPT; ++j)
            if (z[j] > tau) { s += z[j]; c += 1.0f; }
        block_reduce2(s, c, sred, tid);
        tau = (s - 1.0f) / c;                     // support shrinks monotonically
        if (c == prevc) break;                    // support stable -> tau is fixed point
        prevc = c;
    }

    // ---- p = clip(z - tau); normalize; threshold; renormalize ----
    float p[EPT];
    float ps = 0.0f, d0 = 0.0f;
#pragma unroll
    for (int j = 0; j < EPT; ++j) { p[j] = fmaxf(z[j] - tau, 0.0f); ps += p[j]; }
    block_reduce2(ps, d0, sred, tid);
    const float inv = 1.0f / fmaxf(ps, 1e-12f);

    float ws = 0.0f, d1 = 0.0f;
#pragma unroll
    for (int j = 0; j < EPT; ++j) {
        float w = p[j] * inv;
        w = (w < 1e-6f) ? 0.0f : w;               // MIN_WEIGHT threshold
        p[j] = w; ws += w;
    }
    block_reduce2(ws, d1, sred, tid);
    const float inv2 = 1.0f / fmaxf(ws, 1e-12f);

    v4f o0, o1;
#pragma unroll
    for (int j = 0; j < 4; ++j) { o0[j] = p[j] * inv2; o1[j] = p[j + 4] * inv2; }
    __builtin_nontemporal_store(o0, (v4f*)(out + base + tid * EPT));
    __builtin_nontemporal_store(o1, (v4f*)(out + base + tid * EPT + 4));
}

extern "C" void kernel_launch(void* const* d_in, const int* in_sizes, int n_in,
                              void* d_out, int out_size, void* d_ws, size_t ws_size,
                              hipStream_t stream) {
    const float*   logits = (const float*)d_in[0];
    const uint8_t* mask   = (const uint8_t*)d_in[1];
    float*         out    = (float*)d_out;
    const int rows = in_sizes[0] / NCOLS;         // B = 16384
    sparsemax_rows<<<rows, TPB, 0, stream>>>(logits, mask, out);
}